// DirectionAssigned_29454885716034
// MI455X (gfx1250) — compile-verified
//
#include <hip/hip_runtime.h>

// out[b,i,j] = x[b,i,j] - (i>=2 ? x[b,i-2,j] : 0)   on 32 x 1 x 1024 x 1024 fp32
//
// Formulated per 16x16 tile as out_t = D*X_t + L*X_{t-1} with banded D, L,
// executed as 5 chained V_WMMA_F32_16X16X4_F32 per tile. Memory-bound:
// ~256MB @ 23.3 TB/s ~= 11us floor; 1.3 GFLOP of f32 WMMA is noise next to that.

typedef float v2f __attribute__((ext_vector_type(2)));
typedef float v8f __attribute__((ext_vector_type(8)));

#define DIM_H 1024
#define DIM_W 1024
#define N_IMG 32
#define TILES_X (DIM_W / 16)                 // 64 column tiles
#define TILES_Y (DIM_H / 16)                 // 64 row tiles
#define TILES_PER_IMG (TILES_X * TILES_Y)    // 4096
#define WAVES_PER_BLOCK 8
#define TOTAL_TILES (N_IMG * TILES_PER_IMG)  // 131072

__global__ __launch_bounds__(256) void shiftsub_wmma_kernel(
    const float* __restrict__ x, float* __restrict__ out)
{
    const int lane   = threadIdx.x & 31;
    const int waveId = threadIdx.x >> 5;
    const int tile   = blockIdx.x * WAVES_PER_BLOCK + waveId;

    const int img = tile / TILES_PER_IMG;
    const int tin = tile - img * TILES_PER_IMG;
    const int rt  = tin / TILES_X;            // row-tile index within image
    const int ct  = tin - rt * TILES_X;       // col-tile index within image

    const int n  = lane & 15;                 // N (column) index of this lane
    const int hi = lane >> 4;                 // half-wave select (K / M split)
    const int m  = n;                         // M index for A-matrix entries

    const long base = (long)img * (long)(DIM_H * DIM_W);
    const int  r0   = rt * 16;
    const int  col  = ct * 16 + n;
    const float* xc = x + base + col;

    v8f acc = {0.f, 0.f, 0.f, 0.f, 0.f, 0.f, 0.f, 0.f};

    // D * X_t : 4 K-slices. A-matrix synthesized from lane id:
    //   D[m][k] = (k==m ? +1 : (k==m-2 ? -1 : 0))
    // B layout mirrors A: low lanes hold K = 4s,4s+1 ; high lanes K = 4s+2,4s+3,
    // each lane's column is n -> each B element is x[r0+k][col].
#pragma unroll
    for (int s = 0; s < 4; ++s) {
        const int kx = 4 * s + 2 * hi;
        v2f a, b;
        a.x = (kx     == m) ? 1.f : ((kx + 2 == m) ? -1.f : 0.f);
        a.y = (kx + 1 == m) ? 1.f : ((kx + 3 == m) ? -1.f : 0.f);
        b.x = xc[(long)(r0 + kx)     * DIM_W];
        b.y = xc[(long)(r0 + kx + 1) * DIM_W];
        acc = __builtin_amdgcn_wmma_f32_16x16x4_f32(
            /*neg_a=*/false, a, /*neg_b=*/false, b,
            /*c_mod=*/(short)0, acc, /*reuse_a=*/false, /*reuse_b=*/false);
    }

    // L * X_{t-1} : top-two-row fix from previous 16-row block.
    // K-slice covers block-local K=12..15 of X_{t-1} = rows r0-4..r0-1;
    // only K=14 (row r0-2) and K=15 (row r0-1) carry -1 (at M=0, M=1).
    // Unconditional WMMA (A == 0 when rt == 0) so EXEC is all-ones at the op.
    {
        const bool prev = (rt > 0);
        v2f a, b;
        a.x = (prev && hi && m == 0) ? -1.f : 0.f;   // K=14
        a.y = (prev && hi && m == 1) ? -1.f : 0.f;   // K=15
        float px = 0.f, py = 0.f;
        if (prev && hi) {                             // divergent loads only
            px = xc[(long)(r0 - 2) * DIM_W];
            py = xc[(long)(r0 - 1) * DIM_W];
        }
        b.x = px;
        b.y = py;
        acc = __builtin_amdgcn_wmma_f32_16x16x4_f32(
            false, a, false, b, (short)0, acc, false, false);
    }

    // C/D layout: lane n = column; low lanes rows 0..7 in v0..7, high lanes 8..15.
    // Output is never re-read -> non-temporal stores (keep L2 for input row reuse).
    float* oc = out + base + col;
#pragma unroll
    for (int v = 0; v < 8; ++v) {
        __builtin_nontemporal_store(acc[v], &oc[(long)(r0 + v + 8 * hi) * DIM_W]);
    }
}

extern "C" void kernel_launch(void* const* d_in, const int* in_sizes, int n_in,
                              void* d_out, int out_size, void* d_ws, size_t ws_size,
                              hipStream_t stream) {
    (void)in_sizes; (void)n_in; (void)d_ws; (void)ws_size; (void)out_size;
    const float* x = (const float*)d_in[0];
    float* out     = (float*)d_out;

    const int blocks = TOTAL_TILES / WAVES_PER_BLOCK;  // 16384 blocks x 256 thr
    shiftsub_wmma_kernel<<<blocks, 256, 0, stream>>>(x, out);
}